// Language_Attention_Relational_Cxt_44753559225076
// MI455X (gfx1250) — compile-verified
//
#include <hip/hip_runtime.h>
#include <math.h>

// CDNA5 (gfx1250) fp32 WMMA fragment types
typedef __attribute__((ext_vector_type(2))) float v2f;
typedef __attribute__((ext_vector_type(8))) float v8f;

#define B_ 64
#define L_ 512
#define H_ 1024

// Branch-free tanh: avoids libm's exec-mask-diverging two-path expansion.
// Clamp keeps exp finite; beyond |x|=12 fp32 tanh is exactly +-1 anyway.
__device__ __forceinline__ float fast_tanh(float x) {
  const float xc = fminf(fmaxf(x, -12.f), 12.f);
  const float e = __expf(2.f * xc);
  return (e - 1.f) / (e + 1.f);
}

// ---------------------------------------------------------------------------
// Accumulate a 16x16 fp32 tile of A (MxK, row-major, row stride lda) * W
// (KxN, row-major, ldb) into c with V_WMMA_F32_16X16X4_F32.
// Straight K run: no per-iteration selects, addresses are pure increments.
// Fragment layouts per CDNA5 ISA 7.12.2:
//   A 16x4 : lanes 0-15 -> M=lane, {K0,K1}; lanes 16-31 -> M=lane-16, {K2,K3}
//   B 4x16 : lanes 0-15 -> N=lane, {K0,K1}; lanes 16-31 -> N=lane-16, {K2,K3}
//   C/D    : vgpr i, lanes 0-15 -> (M=i, N=lane); lanes 16-31 -> (M=i+8, N=lane-16)
// ---------------------------------------------------------------------------
__device__ __forceinline__ v8f wmma_gemm_part(const float* __restrict__ A,
                                              const float* __restrict__ W,
                                              int K, int lda, int ldb,
                                              int mBase, int nBase, v8f c) {
  const int lane  = threadIdx.x & 31;
  const int khalf = (lane >> 4) * 2;  // 0 for lanes 0-15, 2 for lanes 16-31
  const float* Arow = A + (size_t)(mBase + (lane & 15)) * lda + khalf;
  const float* Bcol = W + (size_t)khalf * ldb + nBase + (lane & 15);
  for (int k0 = 0; k0 < K; k0 += 4) {
    v2f a = *(const v2f*)(Arow + k0);               // one global_load_b64
    v2f b;
    b.x = Bcol[(size_t)k0 * ldb];
    b.y = Bcol[(size_t)k0 * ldb + ldb];
    c = __builtin_amdgcn_wmma_f32_16x16x4_f32(false, a, false, b, (short)0, c,
                                              false, false);
  }
  return c;
}

// ---------------------------------------------------------------------------
// Stage 1: h_r = tanh(concat(hA, hB) @ W_r)   [3][B][H]
// grid (H/16, B/16, 3), block 32 (one wave per 16x16 tile; EXEC all-ones)
// Concat handled as two straight K runs: A0 against W[0:H), A1 against W[H:2H).
// ---------------------------------------------------------------------------
__global__ void relational_kernel(const float* __restrict__ hs,
                                  const float* __restrict__ ho,
                                  const float* __restrict__ ha,
                                  const float* __restrict__ Was,
                                  const float* __restrict__ Wso,
                                  const float* __restrict__ Woa,
                                  float* __restrict__ hws) {
  const int r = blockIdx.z;
  const float *A0, *A1, *W;
  if (r == 0)      { A0 = ha; A1 = hs; W = Was; }   // h_as = tanh([h_a,h_s]W_as)
  else if (r == 1) { A0 = hs; A1 = ho; W = Wso; }   // h_so
  else             { A0 = ho; A1 = ha; W = Woa; }   // h_oa
  const int nBase = blockIdx.x * 16, mBase = blockIdx.y * 16;
  v8f c = {0.f, 0.f, 0.f, 0.f, 0.f, 0.f, 0.f, 0.f};
  c = wmma_gemm_part(A0, W,                    H_, H_, H_, mBase, nBase, c);
  c = wmma_gemm_part(A1, W + (size_t)H_ * H_,  H_, H_, H_, mBase, nBase, c);
  const int lane = threadIdx.x & 31;
  const int n = lane & 15, mOff = (lane >> 4) * 8;
  float* out = hws + (size_t)r * B_ * H_;
  for (int i = 0; i < 8; ++i)
    out[(size_t)(mBase + mOff + i) * H_ + nBase + n] = fast_tanh(c[i]);
}

// ---------------------------------------------------------------------------
// Stage 2: target_r = h_r @ Win_r   [3][B][H]
// ---------------------------------------------------------------------------
__global__ void target_kernel(const float* __restrict__ hws,
                              const float* __restrict__ Win_as,
                              const float* __restrict__ Win_so,
                              const float* __restrict__ Win_oa,
                              float* __restrict__ tws) {
  const int r = blockIdx.z;
  const float* W = (r == 0) ? Win_as : (r == 1 ? Win_so : Win_oa);
  const float* A = hws + (size_t)r * B_ * H_;
  const int nBase = blockIdx.x * 16, mBase = blockIdx.y * 16;
  v8f c = {0.f, 0.f, 0.f, 0.f, 0.f, 0.f, 0.f, 0.f};
  c = wmma_gemm_part(A, W, H_, H_, H_, mBase, nBase, c);
  const int lane = threadIdx.x & 31;
  const int n = lane & 15, mOff = (lane >> 4) * 8;
  float* out = tws + (size_t)r * B_ * H_;
  for (int i = 0; i < 8; ++i)
    out[(size_t)(mBase + mOff + i) * H_ + nBase + n] = c[i];
}

// ---------------------------------------------------------------------------
// Stage 3 (fused over all 3 relations): one pass over enc_out computes
// scores for all 3 targets + softmax.  One block per batch, 8 wave32s.
// Each wave owns score rows l = w, w+8, ...; lanes split H and reduce via
// wave32 shfl_xor.  enc_out (134 MB) read ONCE for all 3 relations.
// ---------------------------------------------------------------------------
__global__ void scores_softmax_kernel(const float* __restrict__ enc,
                                      const float* __restrict__ tws,
                                      float* __restrict__ pws) {
  __shared__ float tgt[3 * H_];   // 12 KB: the 3 target vectors for this batch
  __shared__ float sc[3 * L_];    // 6 KB: raw scores
  __shared__ float red[256];      // reduction scratch
  const int b = blockIdx.x, t = threadIdx.x;
  for (int i = t; i < 3 * H_; i += 256) {
    const int r = i >> 10, h = i & (H_ - 1);
    tgt[i] = tws[((size_t)r * B_ + b) * H_ + h];
  }
  __syncthreads();

  const int w = t >> 5, ln = t & 31;        // 8 waves of 32 (wave32)
  const float* t0 = tgt;
  const float* t1 = tgt + H_;
  const float* t2 = tgt + 2 * H_;
  for (int l = w; l < L_; l += 8) {
    const float* row = enc + ((size_t)b * L_ + l) * H_;
    float a0 = 0.f, a1 = 0.f, a2 = 0.f;
    for (int h = ln * 4; h < H_; h += 128) {   // lanes read consecutive float4
      const float4 e = *(const float4*)(row + h);
      a0 += e.x * t0[h] + e.y * t0[h + 1] + e.z * t0[h + 2] + e.w * t0[h + 3];
      a1 += e.x * t1[h] + e.y * t1[h + 1] + e.z * t1[h + 2] + e.w * t1[h + 3];
      a2 += e.x * t2[h] + e.y * t2[h + 1] + e.z * t2[h + 2] + e.w * t2[h + 3];
    }
    for (int off = 16; off > 0; off >>= 1) {   // wave32 reduction
      a0 += __shfl_xor(a0, off, 32);
      a1 += __shfl_xor(a1, off, 32);
      a2 += __shfl_xor(a2, off, 32);
    }
    if (ln == 0) { sc[l] = a0; sc[L_ + l] = a1; sc[2 * L_ + l] = a2; }
  }
  __syncthreads();

  // softmax over L=512 per relation (each thread owns elements t and t+256)
  for (int r = 0; r < 3; ++r) {
    const float s0 = sc[r * L_ + t], s1 = sc[r * L_ + t + 256];
    red[t] = fmaxf(s0, s1);
    __syncthreads();
    for (int s = 128; s > 0; s >>= 1) {
      if (t < s) red[t] = fmaxf(red[t], red[t + s]);
      __syncthreads();
    }
    const float mx = red[0];
    __syncthreads();
    const float e0 = __expf(s0 - mx), e1 = __expf(s1 - mx);
    red[t] = e0 + e1;
    __syncthreads();
    for (int s = 128; s > 0; s >>= 1) {
      if (t < s) red[t] += red[t + s];
      __syncthreads();
    }
    const float inv = 1.f / red[0];
    __syncthreads();
    float* pr = pws + ((size_t)r * B_ + b) * L_;
    pr[t] = e0 * inv;
    pr[t + 256] = e1 * inv;
  }
}

// ---------------------------------------------------------------------------
// Stage 4 (fused over all 3 relations): weighted_r[b,h] = sum_l p_r[l]*enc[b,l,h]
// Second (final) pass over enc_out.  One block per batch; thread t owns
// columns h = 4t..4t+3, so each l-iteration reads one full 4 KB row coalesced.
// ---------------------------------------------------------------------------
__global__ void weighted_ctx_kernel(const float* __restrict__ enc,
                                    const float* __restrict__ pws,
                                    float* __restrict__ wws) {
  __shared__ float pl[3 * L_];   // 6 KB
  const int b = blockIdx.x, t = threadIdx.x;
  for (int i = t; i < 3 * L_; i += 256) {
    const int r = i >> 9, l = i & (L_ - 1);
    pl[i] = pws[((size_t)r * B_ + b) * L_ + l];
  }
  __syncthreads();
  const int h0 = t * 4;
  float4 a0 = {0, 0, 0, 0}, a1 = {0, 0, 0, 0}, a2 = {0, 0, 0, 0};
  for (int l = 0; l < L_; ++l) {
    const float4 e = *(const float4*)(enc + ((size_t)b * L_ + l) * H_ + h0);
    const float s0 = pl[l], s1 = pl[L_ + l], s2 = pl[2 * L_ + l];
    a0.x += s0 * e.x; a0.y += s0 * e.y; a0.z += s0 * e.z; a0.w += s0 * e.w;
    a1.x += s1 * e.x; a1.y += s1 * e.y; a1.z += s1 * e.z; a1.w += s1 * e.w;
    a2.x += s2 * e.x; a2.y += s2 * e.y; a2.z += s2 * e.z; a2.w += s2 * e.w;
  }
  *(float4*)(wws + ((size_t)0 * B_ + b) * H_ + h0) = a0;
  *(float4*)(wws + ((size_t)1 * B_ + b) * H_ + h0) = a1;
  *(float4*)(wws + ((size_t)2 * B_ + b) * H_ + h0) = a2;
}

// ---------------------------------------------------------------------------
// Stage 5: sa_r = tanh(concat(weighted_r, h_r) @ Wout_r), broadcast to [B,F,H].
// tanh + F-way broadcast fused into the WMMA epilogue.  F is read from the
// device-side scalar (no host sync -> graph-capture safe).
// ---------------------------------------------------------------------------
__global__ void output_kernel(const float* __restrict__ wws,
                              const float* __restrict__ hws,
                              const float* __restrict__ Wout_as,
                              const float* __restrict__ Wout_so,
                              const float* __restrict__ Wout_oa,
                              const int* __restrict__ dF,
                              float* __restrict__ out) {
  const int r = blockIdx.z;
  const float* W = (r == 0) ? Wout_as : (r == 1 ? Wout_so : Wout_oa);
  const float* A0 = wws + (size_t)r * B_ * H_;   // weighted_ctx
  const float* A1 = hws + (size_t)r * B_ * H_;   // h_r
  const int nBase = blockIdx.x * 16, mBase = blockIdx.y * 16;
  v8f c = {0.f, 0.f, 0.f, 0.f, 0.f, 0.f, 0.f, 0.f};
  c = wmma_gemm_part(A0, W,                   H_, H_, H_, mBase, nBase, c);
  c = wmma_gemm_part(A1, W + (size_t)H_ * H_, H_, H_, H_, mBase, nBase, c);
  const int F = *dF;
  const int lane = threadIdx.x & 31;
  const int n = lane & 15, mOff = (lane >> 4) * 8;
  float* outr = out + (size_t)r * B_ * (size_t)F * H_;
  for (int i = 0; i < 8; ++i) {
    const float v = fast_tanh(c[i]);
    const size_t base = (size_t)(mBase + mOff + i) * F * H_ + nBase + n;
    for (int f = 0; f < F; ++f) outr[base + (size_t)f * H_] = v;
  }
}

// ---------------------------------------------------------------------------
extern "C" void kernel_launch(void* const* d_in, const int* in_sizes, int n_in,
                              void* d_out, int out_size, void* d_ws, size_t ws_size,
                              hipStream_t stream) {
  (void)in_sizes; (void)n_in; (void)out_size; (void)ws_size;
  const float* hs      = (const float*)d_in[0];
  const float* ho      = (const float*)d_in[1];
  const float* ha      = (const float*)d_in[2];
  const float* enc     = (const float*)d_in[3];
  const int*   dF      = (const int*)  d_in[4];
  const float* Was     = (const float*)d_in[5];
  const float* Wso     = (const float*)d_in[6];
  const float* Woa     = (const float*)d_in[7];
  const float* Win_as  = (const float*)d_in[8];
  const float* Win_so  = (const float*)d_in[9];
  const float* Win_oa  = (const float*)d_in[10];
  const float* Wout_as = (const float*)d_in[11];
  const float* Wout_so = (const float*)d_in[12];
  const float* Wout_oa = (const float*)d_in[13];

  // workspace layout (floats): h_r | target_r | p_r | weighted_r  (~2.75 MB)
  float* ws  = (float*)d_ws;
  float* hws = ws;
  float* tws = hws + 3 * B_ * H_;
  float* pws = tws + 3 * B_ * H_;
  float* wws = pws + 3 * B_ * L_;

  const dim3 gTile(H_ / 16, B_ / 16, 3);   // one wave32 per 16x16 tile
  relational_kernel<<<gTile, 32, 0, stream>>>(hs, ho, ha, Was, Wso, Woa, hws);
  target_kernel<<<gTile, 32, 0, stream>>>(hws, Win_as, Win_so, Win_oa, tws);
  scores_softmax_kernel<<<B_, 256, 0, stream>>>(enc, tws, pws);
  weighted_ctx_kernel<<<B_, 256, 0, stream>>>(enc, pws, wws);
  output_kernel<<<gTile, 32, 0, stream>>>(wws, hws, Wout_as, Wout_so, Wout_oa,
                                          dF, (float*)d_out);
}